// AttFusion_56805237457110
// MI455X (gfx1250) — compile-verified
//
#include <hip/hip_runtime.h>
#include <hip/hip_bf16.h>
#include <stdint.h>

// Problem constants (from reference): x is (5, 512, 128, 256) fp32.
#define CAV   5
#define CDIM  512
#define HW    32768          // 128*256
#define TILE  16             // spatial positions per block
#define ROWS  (CAV * CDIM)   // 2560 (k,c) rows
#define LDS_F4 (ROWS * (TILE / 4))   // 10240 float4 = 160 KB token tile

typedef __attribute__((ext_vector_type(2))) float v2f;
typedef __attribute__((ext_vector_type(4))) float f4;
typedef __attribute__((ext_vector_type(8))) float v8f;
// Matches the builtin's parameter type from the round-1 diagnostic:
// "int __vector_size__(4*sizeof(int)) __device__ *"  (AS1 = __device__)
typedef __attribute__((__vector_size__(4 * sizeof(int)))) int v4i;

#ifndef __has_builtin
#define __has_builtin(x) 0
#endif

#if __has_builtin(__builtin_amdgcn_global_load_async_to_lds_b128)
#define HAVE_ASYNC 1
#else
#define HAVE_ASYNC 0
#endif

#if __has_builtin(__builtin_amdgcn_wmma_f32_16x16x4_f32)
#define HAVE_WMMA4 1
#else
#define HAVE_WMMA4 0
#endif

__device__ __forceinline__ float v8_get(v8f v, int i) {
  switch (i) {
    case 0: return v[0]; case 1: return v[1]; case 2: return v[2]; case 3: return v[3];
    case 4: return v[4]; case 5: return v[5]; case 6: return v[6]; default: return v[7];
  }
}

__global__ __launch_bounds__(256) void AttFusion_kernel(const float* __restrict__ x,
                                                        const int* __restrict__ rlp,
                                                        float* __restrict__ out) {
  // 160 KB token tile + score/weight scratch (CDNA5 WGP: 320 KB LDS).
  __shared__ f4    tok4[LDS_F4];               // [row = k*512+c][4 x float4 over 16 b]
  __shared__ float scoresLDS[CAV * TILE];
  __shared__ float weightsLDS[CAV * TILE];

  const int tid = threadIdx.x;
  const int b0  = blockIdx.x * TILE;           // 16-aligned -> 64B-aligned float4
  const f4* xg4 = (const f4*)x;

  // ---- Phase A: single-pass HBM -> LDS stage of the whole 5x512x16 tile ----
  // 10240 float4 chunks, 40 per thread; consecutive lanes hit consecutive 16B
  // chunks of the same (k,c) row (rows are 64B in memory for 16 consecutive b).
#pragma unroll 4
  for (int i = 0; i < LDS_F4 / 256; ++i) {
    const int idx = i * 256 + tid;             // float4 id == LDS float4 index
    const int row = idx >> 2;                  // k*512 + c
    const int q   = idx & 3;
    const f4* gp  = xg4 + (size_t)row * (HW / 4) + (b0 >> 2) + q;
#if HAVE_ASYNC
    __builtin_amdgcn_global_load_async_to_lds_b128(
        (__attribute__((address_space(1))) v4i*)gp,
        (__attribute__((address_space(3))) v4i*)&tok4[idx],
        0, 0);
#else
    tok4[idx] = *gp;
#endif
  }
  if (tid < CAV * TILE) scoresLDS[tid] = 0.0f;
#if HAVE_ASYNC
#if __has_builtin(__builtin_amdgcn_s_wait_asynccnt)
  __builtin_amdgcn_s_wait_asynccnt(0);
#else
  asm volatile("s_wait_asynccnt 0" ::: "memory");
#endif
#endif
  __syncthreads();

  const float* tok = (const float*)tok4;       // tok[(k*512+c)*16 + b_local]

  // ---- Phase B: scores s[k][b] = <T0(b), Tk(b)> via WMMA f32 16x16x4 ----
  // Per wave: 64-channel slice; A = T0 tile (16 pos x 4 ch), B = Tk tile
  // (4 ch x 16 pos); Gram diagonal = per-position dots. FLOP waste is free
  // (0.34 GFLOP total problem vs multi-TFLOP WMMA ceiling).
#if HAVE_WMMA4
  {
    const int lane = tid & 31;
    const int wave = tid >> 5;                 // 0..7 -> channel slice [wave*64, +64)
    const int m    = lane & 15;                // position / N index
    const int kh   = (lane >> 4) << 1;         // K half: lanes 16-31 hold K=2,3
    v8f acc[CAV];
#pragma unroll
    for (int k = 0; k < CAV; ++k) acc[k] = (v8f){0.f,0.f,0.f,0.f,0.f,0.f,0.f,0.f};
    const int cbase = wave * 64;
#pragma unroll 4
    for (int s = 0; s < 16; ++s) {
      const int cc = cbase + s * 4 + kh;
      v2f a;
      a.x = tok[(0 * CDIM + cc) * TILE + m];
      a.y = tok[(0 * CDIM + cc + 1) * TILE + m];
#pragma unroll
      for (int k = 0; k < CAV; ++k) {
        v2f bf;
        bf.x = tok[(k * CDIM + cc) * TILE + m];
        bf.y = tok[(k * CDIM + cc + 1) * TILE + m];
        acc[k] = __builtin_amdgcn_wmma_f32_16x16x4_f32(
            false, a, false, bf, (short)0, acc[k], false, false);
      }
    }
    // Diagonal extraction (C/D layout: VGPR r holds M=r in lanes 0-15 and
    // M=8+r in lanes 16-31): diag i<8 -> lane i, VGPR i; diag i>=8 -> lane i+16,
    // VGPR i-8. Reduce the 8 per-wave partials with ds_add_f32.
    const bool lo = lane < 8, hi = lane >= 24;
    if (lo || hi) {
      const int pos = lo ? lane : (lane - 16);
      const int r   = lo ? lane : (lane - 24);
#pragma unroll
      for (int k = 0; k < CAV; ++k)
        atomicAdd(&scoresLDS[k * TILE + pos], v8_get(acc[k], r));
    }
  }
#else
  {
    const int pos = tid & 15;
    const int c0  = (tid >> 4) * 32;
    float p[CAV] = {0.f, 0.f, 0.f, 0.f, 0.f};
    for (int c = c0; c < c0 + 32; ++c) {
      const float t0 = tok[(0 * CDIM + c) * TILE + pos];
#pragma unroll
      for (int k = 0; k < CAV; ++k) p[k] += t0 * tok[(k * CDIM + c) * TILE + pos];
    }
#pragma unroll
    for (int k = 0; k < CAV; ++k) atomicAdd(&scoresLDS[k * TILE + pos], p[k]);
  }
#endif
  __syncthreads();

  // ---- softmax over k (masked k >= record_len), query 0 only ----
  if (tid < TILE) {
    int rl = rlp[0];
    rl = rl < 0 ? 0 : (rl > CAV ? CAV : rl);
    const float scale = 0.044194173824159216f;   // 1/sqrt(512)
    float wv[CAV] = {0.f, 0.f, 0.f, 0.f, 0.f};
    if (rl > 0) {                                 // rl==0 -> nan_to_num -> zeros
      float s[CAV];
      float mx = -3.4e38f;
      for (int k = 0; k < rl; ++k) {
        s[k] = scoresLDS[k * TILE + tid] * scale;
        mx = fmaxf(mx, s[k]);
      }
      float sum = 0.f;
      for (int k = 0; k < rl; ++k) { wv[k] = __expf(s[k] - mx); sum += wv[k]; }
      const float inv = 1.0f / sum;
      for (int k = 0; k < rl; ++k) wv[k] *= inv;
    }
#pragma unroll
    for (int k = 0; k < CAV; ++k) weightsLDS[k * TILE + tid] = wv[k];
  }
  __syncthreads();

  // ---- Phase C: out[c][b] = sum_k w[k][b] * Tk[c][b], float4 over b ----
  {
    const int q    = tid & 3;                   // which float4 of the 16-b row
    const int crow = tid >> 2;                  // 0..63, loop adds 64
    f4 w4[CAV];
#pragma unroll
    for (int k = 0; k < CAV; ++k) {
      w4[k].x = weightsLDS[k * TILE + q * 4 + 0];
      w4[k].y = weightsLDS[k * TILE + q * 4 + 1];
      w4[k].z = weightsLDS[k * TILE + q * 4 + 2];
      w4[k].w = weightsLDS[k * TILE + q * 4 + 3];
    }
    f4* og4 = (f4*)out;
#pragma unroll
    for (int i = 0; i < 8; ++i) {
      const int c = crow + i * 64;
      f4 acc = (f4){0.f, 0.f, 0.f, 0.f};
#pragma unroll
      for (int k = 0; k < CAV; ++k) {
        const f4 t = tok4[(k * CDIM + c) * 4 + q];
        acc += w4[k] * t;
      }
      og4[(size_t)c * (HW / 4) + (b0 >> 2) + q] = acc;
    }
  }
}

extern "C" void kernel_launch(void* const* d_in, const int* in_sizes, int n_in,
                              void* d_out, int out_size, void* d_ws, size_t ws_size,
                              hipStream_t stream) {
  const float* x  = (const float*)d_in[0];
  const int*   rl = (const int*)d_in[1];
  float*       o  = (float*)d_out;
  dim3 grid(HW / TILE);   // 2048 blocks
  dim3 block(256);        // 8 wave32
  AttFusion_kernel<<<grid, block, 0, stream>>>(x, rl, o);
}